// MinibatchDiscrimination_31404800868854
// MI455X (gfx1250) — compile-verified
//
#include <hip/hip_runtime.h>
#include <hip/hip_bf16.h>
#include <cstdint>

// Problem sizes (fixed by the reference)
#define BB   256   // batch
#define FF   1024  // features
#define NKK  100   // number of kernels
#define KDD  50    // kernel dim
#define PAD  52    // padded row stride (208 B = 13*16 B -> float4-aligned rows)

typedef __attribute__((ext_vector_type(2))) float v2f;
typedef __attribute__((ext_vector_type(8))) float v8f;
typedef __attribute__((ext_vector_type(4))) int   v4i;

#if __has_builtin(__builtin_amdgcn_global_load_async_to_lds_b128) && \
    __has_builtin(__builtin_amdgcn_s_wait_asynccnt)
#define HAVE_ASYNC_LDS 1
typedef __attribute__((address_space(1))) v4i g_v4i;   // global (AS1) 16B vector
typedef __attribute__((address_space(3))) v4i l_v4i;   // LDS    (AS3) 16B vector
#endif

// ---------------------------------------------------------------------------
// Kernel A: act[k][m][d] = sum_f x[m][f] * kernel[k][f][d], act row stride PAD.
// One wave per (mtile, k); wave computes a 16x64 C tile (cols >= 50 clamped,
// never stored) via V_WMMA_F32_16X16X4_F32, K-stepped by 4 over F = 1024.
//
// A-frag (32-bit 16x4): lanes 0-15 hold row M=lane,    K = f+0, f+1
//                       lanes 16-31 hold row M=lane-16, K = f+2, f+3
// B-frag (32-bit 4x16): lanes 0-15 col N=lane,    K = f+0 (v.x), f+1 (v.y)
//                       lanes 16-31 col N=lane-16, K = f+2 (v.x), f+3 (v.y)
// C/D: VGPR r -> M = r (lanes 0-15) / r+8 (lanes 16-31), N = lane % 16
// ---------------------------------------------------------------------------
__global__ __launch_bounds__(256) void mbd_gemm(const float* __restrict__ x,
                                                const float* __restrict__ w,
                                                float* __restrict__ act) {
  const int lane   = threadIdx.x & 31;
  const int waveId = blockIdx.x * 8 + (threadIdx.x >> 5);  // 0..1599
  const int k      = waveId >> 4;   // 0..99
  const int mtile  = waveId & 15;   // 0..15
  const int half   = lane >> 4;     // 0: lanes 0-15, 1: lanes 16-31
  const int l      = lane & 15;

  const float* xrow = x + (size_t)(mtile * 16 + l) * FF + 2 * half;   // 8B aligned
  const float* kb   = w + (size_t)k * FF * KDD + (size_t)(2 * half) * KDD;

  int ncol[4];
  const float* bptr[4];
#pragma unroll
  for (int j = 0; j < 4; ++j) {
    int n = j * 16 + l;
    ncol[j] = n;
    int nc = (n < KDD) ? n : (KDD - 1);  // clamp in-bounds; junk cols never stored
    bptr[j] = kb + nc;
  }

  v8f c0 = {}, c1 = {}, c2 = {}, c3 = {};

#pragma unroll 2
  for (int f = 0; f < FF; f += 4) {
    v2f a = *(const v2f*)(xrow + f);
    const size_t o = (size_t)f * KDD;
    v2f b0 = { bptr[0][o], bptr[0][o + KDD] };
    v2f b1 = { bptr[1][o], bptr[1][o + KDD] };
    v2f b2 = { bptr[2][o], bptr[2][o + KDD] };
    v2f b3 = { bptr[3][o], bptr[3][o + KDD] };
    c0 = __builtin_amdgcn_wmma_f32_16x16x4_f32(false, a, false, b0, (short)0, c0, false, false);
    c1 = __builtin_amdgcn_wmma_f32_16x16x4_f32(false, a, false, b1, (short)0, c1, false, false);
    c2 = __builtin_amdgcn_wmma_f32_16x16x4_f32(false, a, false, b2, (short)0, c2, false, false);
    c3 = __builtin_amdgcn_wmma_f32_16x16x4_f32(false, a, false, b3, (short)0, c3, false, false);
  }

  float* actk = act + (size_t)k * BB * PAD;
  v8f cs[4] = {c0, c1, c2, c3};
#pragma unroll
  for (int j = 0; j < 4; ++j) {
    const int n = ncol[j];
    if (n < KDD) {
#pragma unroll
      for (int r = 0; r < 8; ++r) {
        const int m = mtile * 16 + r + half * 8;
        actk[(size_t)m * PAD + n] = cs[j][r];
      }
    }
  }
}

// ---------------------------------------------------------------------------
// Kernel B: out[b, FF+k] = sum_{b'} exp(-sum_d |act[b,k,d]-act[b',k,d]|)
// One block per k. act_k (256 x PAD f32 = 53 KB) staged in LDS, preferentially
// via GLOBAL_LOAD_ASYNC_TO_LDS_B128 (ASYNCcnt). Each thread keeps its own row
// in registers as float4 chunks; the b' loop reads LDS at lane-uniform,
// 16B-aligned addresses -> ds_load_b128 broadcasts, conflict free.
// ---------------------------------------------------------------------------
__global__ __launch_bounds__(256) void mbd_pair(const float* __restrict__ act,
                                                float* __restrict__ out) {
  __shared__ float s[BB * PAD];          // 53,248 B of the 320 KB WGP LDS
  const int k   = blockIdx.x;
  const int tid = threadIdx.x;

  const float* actk = act + (size_t)k * BB * PAD;  // contiguous padded block

#if HAVE_ASYNC_LDS
  // 3328 float4s total = 13 per thread; per-lane global->LDS async DMA
#pragma unroll
  for (int v = 0; v < (BB * PAD / 4) / 256; ++v) {
    const int i = (v * 256 + tid) * 4;
    __builtin_amdgcn_global_load_async_to_lds_b128(
        (g_v4i*)(actk + i), (l_v4i*)&s[i], 0, 0);
  }
  __builtin_amdgcn_s_wait_asynccnt(0);
#else
#pragma unroll
  for (int v = 0; v < (BB * PAD / 4) / 256; ++v) {
    const int i = (v * 256 + tid) * 4;
    *(float4*)&s[i] = *(const float4*)(actk + i);
  }
#endif
  __syncthreads();

  // Own row in registers (chunk 12: only .x,.y are real data)
  float4 my4[13];
#pragma unroll
  for (int j = 0; j < 13; ++j) my4[j] = *(const float4*)(s + tid * PAD + j * 4);

  float acc = 0.f;
  for (int bp = 0; bp < BB; ++bp) {
    const float4* row4 = (const float4*)(s + bp * PAD);  // lane-uniform, 16B aligned
    float l1 = 0.f;
#pragma unroll
    for (int j = 0; j < 12; ++j) {
      float4 r = row4[j];
      l1 += fabsf(my4[j].x - r.x) + fabsf(my4[j].y - r.y) +
            fabsf(my4[j].z - r.z) + fabsf(my4[j].w - r.w);
    }
    float2 rt = *(const float2*)(s + bp * PAD + 48);
    l1 += fabsf(my4[12].x - rt.x) + fabsf(my4[12].y - rt.y);
    acc += __expf(-l1);
  }
  out[(size_t)tid * (FF + NKK) + FF + k] = acc;
}

// ---------------------------------------------------------------------------
// Kernel C: out[b, 0:FF] = x[b, :]  (float4 copies; row pitch 1124 floats =
// 4496 B = 281 * 16 B, so both sides stay 16B aligned)
// ---------------------------------------------------------------------------
__global__ __launch_bounds__(256) void mbd_copy(const float* __restrict__ x,
                                                float* __restrict__ out) {
  const int idx = blockIdx.x * blockDim.x + threadIdx.x;  // 0 .. BB*FF/4-1
  const int b  = idx >> 8;          // FF/4 = 256 float4 per row
  const int f4 = idx & 255;
  const float4 v = ((const float4*)(x + (size_t)b * FF))[f4];
  ((float4*)(out + (size_t)b * (FF + NKK)))[f4] = v;
}

extern "C" void kernel_launch(void* const* d_in, const int* in_sizes, int n_in,
                              void* d_out, int out_size, void* d_ws, size_t ws_size,
                              hipStream_t stream) {
  (void)in_sizes; (void)n_in; (void)out_size; (void)ws_size;
  const float* x = (const float*)d_in[0];       // [256, 1024]
  const float* w = (const float*)d_in[1];       // [100, 1024, 50]
  float* out = (float*)d_out;                   // [256, 1124]
  float* act = (float*)d_ws;                    // [100, 256, PAD] = 5.33 MB

  mbd_gemm<<<200, 256, 0, stream>>>(x, w, act);   // 1600 waves
  mbd_pair<<<100, 256, 0, stream>>>(act, out);    // one block per k
  mbd_copy<<<BB * FF / 4 / 256, 256, 0, stream>>>(x, out);
}